// PFFE_5866925326620
// MI455X (gfx1250) — compile-verified
//
#include <hip/hip_runtime.h>
#include <math.h>

#define N_IMG 16
#define N_CH  64
#define H 256
#define W 256
#define HW (H * W)           // 65536 pixels per plane
#define TILE_W 64
#define TILE_H 32
#define HALO_W (TILE_W + 6)  // 70
#define HALO_H (TILE_H + 6)  // 38
#define LDS_PITCH 72

typedef __attribute__((ext_vector_type(2))) float v2f;
typedef __attribute__((ext_vector_type(8))) float v8f;

// 7 fixed kernels, each zero-padded to 7x7, flattened row-major (49 each).
// Order: gx3, gy3, gx5, gy5, gx7, gy7, lap  (matches reference channel order)
__constant__ float FLT[7][49] = {
    // gx3
    {0,0,0,0,0,0,0,  0,0,0,0,0,0,0,  0,0,-1,0,1,0,0,  0,0,-2,0,2,0,0,
     0,0,-1,0,1,0,0,  0,0,0,0,0,0,0,  0,0,0,0,0,0,0},
    // gy3
    {0,0,0,0,0,0,0,  0,0,0,0,0,0,0,  0,0,-1,-2,-1,0,0,  0,0,0,0,0,0,0,
     0,0,1,2,1,0,0,  0,0,0,0,0,0,0,  0,0,0,0,0,0,0},
    // gx5
    {0,0,0,0,0,0,0,  0,-1,-2,0,2,1,0,  0,-4,-8,0,8,4,0,  0,-6,-12,0,12,6,0,
     0,-4,-8,0,8,4,0,  0,-1,-2,0,2,1,0,  0,0,0,0,0,0,0},
    // gy5
    {0,0,0,0,0,0,0,  0,-1,-4,-6,-4,-1,0,  0,-2,-8,-12,-8,-2,0,  0,0,0,0,0,0,0,
     0,2,8,12,8,2,0,  0,1,4,6,4,1,0,  0,0,0,0,0,0,0},
    // gx7
    {-1,-4,-5,0,5,4,1,  -6,-24,-30,0,30,24,6,  -15,-60,-75,0,75,60,15,
     -20,-80,-100,0,100,80,20,  -15,-60,-75,0,75,60,15,  -6,-24,-30,0,30,24,6,
     -1,-4,-5,0,5,4,1},
    // gy7
    {-1,-6,-15,-20,-15,-6,-1,  -4,-24,-60,-80,-60,-24,-4,  -5,-30,-75,-100,-75,-30,-5,
     0,0,0,0,0,0,0,  5,30,75,100,75,30,5,  4,24,60,80,60,24,4,
     1,6,15,20,15,6,1},
    // lap
    {0,0,0,0,0,0,0,  0,0,0,0,0,0,0,  0,0,0,1,0,0,0,  0,0,1,-4,1,0,0,
     0,0,0,1,0,0,0,  0,0,0,0,0,0,0,  0,0,0,0,0,0,0}
};

// ---------- async global->LDS copy (CDNA5 path with safe fallback) ----------
#if defined(__AMDGCN__) && __has_builtin(__builtin_amdgcn_global_load_async_to_lds_b32)
#define USE_ASYNC_LDS 1
#else
#define USE_ASYNC_LDS 0
#endif

typedef __attribute__((address_space(1))) int* glb_i32_ptr;
typedef __attribute__((address_space(3))) int* lds_i32_ptr;

__device__ __forceinline__ void halo_copy(float* dst_lds, const float* src_glb) {
#if USE_ASYNC_LDS
  // int-to-pointer casts sidestep generic<->AS cast restrictions; the builtin
  // takes address-space-qualified int* operands.
  __builtin_amdgcn_global_load_async_to_lds_b32(
      (glb_i32_ptr)(unsigned long long)src_glb,
      (lds_i32_ptr)(unsigned)(unsigned long long)dst_lds,
      0, 0);
#else
  *dst_lds = *src_glb;
#endif
}

__device__ __forceinline__ void wait_async_done() {
#if USE_ASYNC_LDS
#if __has_builtin(__builtin_amdgcn_s_wait_asynccnt)
  __builtin_amdgcn_s_wait_asynccnt(0);
#else
  asm volatile("s_wait_asynccnt 0" ::: "memory");
#endif
#endif
}

// ---------------- stage 1: channel mean -> gray (float4 coalesced) ----------
__global__ __launch_bounds__(256) void k_gray(const float* __restrict__ x,
                                              float* __restrict__ gray) {
  int t = blockIdx.x * 256 + threadIdx.x;      // 262144 threads, one float4 each
  int n = t >> 14;                              // image
  int p = t & 16383;                            // float4 index within plane
  const float4* xb = (const float4*)x + (size_t)n * N_CH * (HW / 4) + p;
  float sx = 0.f, sy = 0.f, sz = 0.f, sw = 0.f;
  for (int c = 0; c < N_CH; ++c) {
    float4 v = xb[(size_t)c * (HW / 4)];
    sx += v.x; sy += v.y; sz += v.z; sw += v.w;
  }
  const float inv = 1.0f / 64.0f;
  float4 o; o.x = sx * inv; o.y = sy * inv; o.z = sz * inv; o.w = sw * inv;
  ((float4*)gray)[(size_t)n * (HW / 4) + p] = o;
}

// -------- stage 2: 7x(7x7) conv as WMMA implicit GEMM + combined + sums -----
// D(16 filters x 16 pixels) = A(16x49 filters, rows>=7 zero) * B(49 x 16 patch)
// via 13 chained V_WMMA_F32_16X16X4_F32 (K padded 49->52).
// B fragments are gathered into registers FIRST (one s_wait_dscnt), then the
// 13 WMMAs issue back-to-back so the matrix pipe streams.
__global__ __launch_bounds__(256) void k_edge_wmma(const float* __restrict__ gray,
                                                   float* __restrict__ comb,
                                                   float* __restrict__ stats) {
  __shared__ float tile[HALO_H * LDS_PITCH];
  __shared__ float red[512];

  const int tid = threadIdx.x;
  const int img = blockIdx.z;
  const int x0 = blockIdx.x * TILE_W;
  const int y0 = blockIdx.y * TILE_H;
  const float* gbase = gray + (size_t)img * HW;

  // zero-fill (borders stay zero = zero padding), then async-load interior
  for (int i = tid; i < HALO_H * LDS_PITCH; i += 256) tile[i] = 0.f;
  __syncthreads();
  for (int i = tid; i < HALO_H * HALO_W; i += 256) {
    int r = i / HALO_W, c = i - r * HALO_W;
    int gy = y0 - 3 + r, gx = x0 - 3 + c;
    if (gy >= 0 && gy < H && gx >= 0 && gx < W)
      halo_copy(&tile[r * LDS_PITCH + c], gbase + gy * W + gx);
  }
  wait_async_done();
  __syncthreads();

  const int lane  = tid & 31;
  const int wv    = tid >> 5;
  const int khalf = lane >> 4;   // ISA: lanes 0-15 hold K 0,1; lanes 16-31 hold K 2,3
  const int mn    = lane & 15;   // filter row (A) == pixel column (B/D)

  // Precompute per-lane A fragments and B LDS offsets for the 13 K-steps.
  float ax[13], ay[13];
  int bo0[13], bo1[13];
#pragma unroll
  for (int t = 0; t < 13; ++t) {
    int k0 = 4 * t + 2 * khalf;
    int k1 = k0 + 1;
    ax[t] = (mn < 7 && k0 < 49) ? FLT[mn][k0] : 0.f;
    ay[t] = (mn < 7 && k1 < 49) ? FLT[mn][k1] : 0.f;
    bo0[t] = (k0 < 49) ? (k0 / 7) * LDS_PITCH + (k0 % 7) : 0;
    bo1[t] = (k1 < 49) ? (k1 / 7) * LDS_PITCH + (k1 % 7) : 0;
  }

  float s1 = 0.f, s2 = 0.f;
  // 128 groups of 16 pixels per tile; 8 waves round-robin (uniform per wave,
  // EXEC all-ones at every WMMA).
  for (int g = wv; g < (TILE_H * TILE_W / 16); g += 8) {
    int py = g >> 2;
    int px = ((g & 3) << 4) + mn;
    int base = py * LDS_PITCH + px;

    // Phase 1: gather all 26 independent B values from LDS.
    float bx[13], by[13];
#pragma unroll
    for (int t = 0; t < 13; ++t) {
      bx[t] = tile[base + bo0[t]];
      by[t] = tile[base + bo1[t]];
    }

    // Phase 2: 13 back-to-back accumulating WMMAs.
    v8f acc = {};
#pragma unroll
    for (int t = 0; t < 13; ++t) {
      v2f a, b;
      a.x = ax[t]; a.y = ay[t];
      b.x = bx[t]; b.y = by[t];
      acc = __builtin_amdgcn_wmma_f32_16x16x4_f32(
          /*neg_a=*/false, a, /*neg_b=*/false, b,
          /*c_mod=*/(short)0, acc, /*reuse_a=*/false, /*reuse_b=*/false);
    }

    // lanes 0-15 hold filters M=0..6 in acc[0..6] for their pixel
    float cmb = (sqrtf(acc[0] * acc[0] + acc[1] * acc[1] + 1e-6f) +
                 sqrtf(acc[2] * acc[2] + acc[3] * acc[3] + 1e-6f) +
                 sqrtf(acc[4] * acc[4] + acc[5] * acc[5] + 1e-6f) +
                 fabsf(acc[6])) * 0.25f;
    if (lane < 16) {
      comb[(size_t)img * HW + (size_t)(y0 + py) * W + (x0 + px)] = cmb;
      s1 += cmb;
      s2 += cmb * cmb;
    }
  }

  // block reduction of (sum, sumsq), then one atomicAdd pair per block
  red[tid] = s1;
  red[256 + tid] = s2;
  __syncthreads();
  for (int s = 128; s > 0; s >>= 1) {
    if (tid < s) {
      red[tid] += red[tid + s];
      red[256 + tid] += red[256 + tid + s];
    }
    __syncthreads();
  }
  if (tid == 0) {
    atomicAdd(&stats[img], red[0]);
    atomicAdd(&stats[16 + img], red[256]);
  }
}

// ---------------- stage 0/3: stats init + finalize ---------------------------
__global__ void k_init(float* stats) {
  if (threadIdx.x < 32) stats[threadIdx.x] = 0.f;
}

__global__ void k_stats(float* stats) {
  int i = threadIdx.x;
  if (i < N_IMG) {
    float s1 = stats[i];
    float s2 = stats[16 + i];
    const float Nf = (float)HW;
    float mu = s1 / Nf;
    float var = (s2 - s1 * mu) / (Nf - 1.0f);  // ddof=1
    var = var > 0.f ? var : 0.f;
    stats[32 + i] = mu;
    stats[48 + i] = 1.0f / (sqrtf(var) + 1e-6f);
  }
}

// ---------------- stage 4: apply x * (1 + sigmoid(5*norm - 2.5)) ------------
__global__ __launch_bounds__(256) void k_apply(const float* __restrict__ x,
                                               const float* __restrict__ comb,
                                               const float* __restrict__ stats,
                                               float* __restrict__ out) {
  int t = blockIdx.x * 256 + threadIdx.x;  // 262144 threads, one float4 column
  int n = t >> 14;
  int p = t & 16383;
  float mu = stats[32 + n];
  float rs = stats[48 + n];

  float4 cb = ((const float4*)comb)[(size_t)n * (HW / 4) + p];
  float fx = 1.0f + 1.0f / (1.0f + expf(-(5.0f * (cb.x - mu) * rs - 2.5f)));
  float fy = 1.0f + 1.0f / (1.0f + expf(-(5.0f * (cb.y - mu) * rs - 2.5f)));
  float fz = 1.0f + 1.0f / (1.0f + expf(-(5.0f * (cb.z - mu) * rs - 2.5f)));
  float fw = 1.0f + 1.0f / (1.0f + expf(-(5.0f * (cb.w - mu) * rs - 2.5f)));

  const float4* xb = (const float4*)x + (size_t)n * N_CH * (HW / 4) + p;
  float4* ob = (float4*)out + (size_t)n * N_CH * (HW / 4) + p;
  for (int c = 0; c < N_CH; ++c) {
    float4 v = xb[(size_t)c * (HW / 4)];
    v.x *= fx; v.y *= fy; v.z *= fz; v.w *= fw;
    ob[(size_t)c * (HW / 4)] = v;
  }
}

extern "C" void kernel_launch(void* const* d_in, const int* in_sizes, int n_in,
                              void* d_out, int out_size, void* d_ws, size_t ws_size,
                              hipStream_t stream) {
  (void)in_sizes; (void)n_in; (void)out_size; (void)ws_size;
  const float* x = (const float*)d_in[0];
  float* out = (float*)d_out;
  float* ws = (float*)d_ws;

  float* gray  = ws;                         // 1,048,576 f32 (4 MB)
  float* comb  = ws + (1 << 20);             // 1,048,576 f32 (4 MB)
  float* stats = ws + (2 << 20);             // 64 f32: S1[16] S2[16] mu[16] rs[16]

  k_init<<<1, 32, 0, stream>>>(stats);
  k_gray<<<(N_IMG * HW / 4) / 256, 256, 0, stream>>>(x, gray);
  dim3 g2(W / TILE_W, H / TILE_H, N_IMG);    // (4, 8, 16)
  k_edge_wmma<<<g2, 256, 0, stream>>>(gray, comb, stats);
  k_stats<<<1, 32, 0, stream>>>(stats);
  k_apply<<<(N_IMG * HW / 4) / 256, 256, 0, stream>>>(x, comb, stats, out);
}